// STGN_14628658610802
// MI455X (gfx1250) — compile-verified
//
#include <hip/hip_runtime.h>
#include <hip/hip_bf16.h>
#include <math.h>

typedef __attribute__((ext_vector_type(2))) float v2f;
typedef __attribute__((ext_vector_type(8))) float v8f;

#define NB 2
#define TT 4
#define NT 8      // NB*TT images
#define C  64
#define H  64
#define Wd 96
#define HW (H*Wd) // 6144
#define K2 9
#define TK 36     // t*k2
#define TKP 48    // padded to 3 WMMA tiles

#if __has_builtin(__builtin_amdgcn_global_load_async_to_lds_b64)
#define HAVE_ASYNC_LDS 1
// builtin wants: (v2i AS(1)* gsrc, v2i AS(3)* ldst, imm offset, imm cpol)
typedef int async_v2i __attribute__((vector_size(8)));
typedef async_v2i __attribute__((address_space(1)))* async_gp;
typedef async_v2i __attribute__((address_space(3)))* async_lp;
#else
#define HAVE_ASYNC_LDS 0
#endif

// ---------------------------------------------------------------------------
// Kernel 1: three 1x1 convs + ReLU as f32 WMMA GEMM.
//   y[a][img][p][o] = relu( sum_c w_a[o][c] * x[img][c][p] )
// One wave computes a 16(o) x 16(p) tile for all three weight matrices
// (shared B loads, 3 accumulators). Output is PIXEL-MAJOR so the attention
// kernel's unfold gathers are contiguous 256B rows.
// ---------------------------------------------------------------------------
__global__ __launch_bounds__(256)
void stgn_conv_kernel(const float* __restrict__ x,
                      const float* __restrict__ w1,
                      const float* __restrict__ w2,
                      const float* __restrict__ w3,
                      float* __restrict__ y)
{
    const int lane = threadIdx.x & 31;
    const int widx = threadIdx.x >> 5;
    const int gw   = blockIdx.x * 8 + widx;        // 12288 waves total
    const int ptile = gw % (HW / 16);
    const int rest  = gw / (HW / 16);
    const int ot    = rest & 3;
    const int img   = rest >> 2;

    const int half = lane >> 4;      // selects K-pair for A/B, M-half for D
    const int ln   = lane & 15;
    const int p0 = ptile * 16;
    const int o0 = ot * 16;

    // weight rows are tiny and reused by every wave -> pull into cache early
    __builtin_prefetch(w1 + (o0 + ln) * C, 0, 3);
    __builtin_prefetch(w2 + (o0 + ln) * C, 0, 3);
    __builtin_prefetch(w3 + (o0 + ln) * C, 0, 3);

    v8f acc0 = {}; v8f acc1 = {}; v8f acc2 = {};
#pragma unroll
    for (int k = 0; k < 16; ++k) {                 // K = 64, 4 per WMMA
        const int c0 = 4 * k + 2 * half;
        // B tile: B[kk][n] = x[img][c0+kk][p0+n]
        v2f b;
        b.x = x[(img * C + c0    ) * HW + p0 + ln];
        b.y = x[(img * C + c0 + 1) * HW + p0 + ln];
        // A tiles: A[m][kk] = w[o0+m][c0+kk]  (row-major -> float2)
        v2f a0 = *(const v2f*)(w1 + (o0 + ln) * C + c0);
        v2f a1 = *(const v2f*)(w2 + (o0 + ln) * C + c0);
        v2f a2 = *(const v2f*)(w3 + (o0 + ln) * C + c0);
        acc0 = __builtin_amdgcn_wmma_f32_16x16x4_f32(false, a0, false, b, (short)0, acc0, false, false);
        acc1 = __builtin_amdgcn_wmma_f32_16x16x4_f32(false, a1, false, b, (short)0, acc1, false, false);
        acc2 = __builtin_amdgcn_wmma_f32_16x16x4_f32(false, a2, false, b, (short)0, acc2, false, false);
    }
    const size_t arrStride = (size_t)NT * HW * C;
    const size_t base = (size_t)(img * HW + p0 + ln) * C;   // pixel-major row
#pragma unroll
    for (int r = 0; r < 8; ++r) {
        const int o = o0 + r + 8 * half;                    // D: M = r + 8*half
        y[                base + o] = fmaxf(acc0[r], 0.0f);
        y[    arrStride + base + o] = fmaxf(acc1[r], 0.0f);
        y[2 * arrStride + base + o] = fmaxf(acc2[r], 0.0f);
    }
}

// ---------------------------------------------------------------------------
// Kernel 2: per-pixel attention. One pixel per 128-thread block (4 waves).
// X1/X2/X3 staged in LDS as 48x64 (rows 36..47 zero). Waves 0..2 each own one
// 16-row M-tile of the WMMA GEMMs; small softmax/mean/var phases are VALU.
// The unfold gather uses gfx1250 async global->LDS copies (ASYNCcnt).
// ---------------------------------------------------------------------------
__global__ __launch_bounds__(128)
void stgn_attn_kernel(const float* __restrict__ y, float* __restrict__ out)
{
    __shared__ __align__(16) float sX1[TKP * C];   // 12 KB
    __shared__ __align__(16) float sX2[TKP * C];   // 12 KB
    __shared__ __align__(16) float sX3[TKP * C];   // 12 KB
    __shared__ __align__(16) float sS1[TKP * TKP]; //  9 KB  score1 (padded)
    __shared__ __align__(16) float sX6[TK * C];    //  9 KB  x5 then x6
    __shared__ float sM1[TT * C], sV1[TT * C], sM2[TT * C], sV2[TT * C];
    __shared__ float sS2[16], sS3[16], sW23[16];
    __shared__ float sDot[TK], sSm4[TK];

    const int tid  = threadIdx.x;
    const int lane = tid & 31;
    const int widx = tid >> 5;
    const int half = lane >> 4;
    const int ln   = lane & 15;

    const int b  = blockIdx.x;          // 0 .. NB*HW-1
    const int nn = b / HW;
    const int p  = b % HW;
    const int hh = p / Wd;
    const int ww = p % Wd;

    // ---- P0: unfold-gather X1/X2/X3 rows into LDS (rows >= 36 zeroed) ----
    // Each row is a contiguous 256B copy: one async B64 op (8B/lane x 32).
    // The in-bounds predicate is wave-uniform (whole wave works on one row).
    for (int row = widx; row < 3 * TKP; row += 4) {
        const int arr = row / TKP;
        const int rr  = row % TKP;
        float* dst = (arr == 0 ? sX1 : (arr == 1 ? sX2 : sX3)) + rr * C;
        bool valid = false;
        const float* src = nullptr;
        if (rr < TK) {
            const int t = rr / K2, k = rr % K2;
            const int h2 = hh + (k / 3) - 1;
            const int w2 = ww + (k % 3) - 1;
            if (h2 >= 0 && h2 < H && w2 >= 0 && w2 < Wd) {
                valid = true;
                src = y + (size_t)arr * NT * HW * C
                        + (size_t)((nn * TT + t) * HW + h2 * Wd + w2) * C;
            }
        }
        if (valid) {
#if HAVE_ASYNC_LDS
            __builtin_amdgcn_global_load_async_to_lds_b64(
                (async_gp)(uintptr_t)(src + 2 * lane),
                (async_lp)(unsigned int)(uintptr_t)(dst + 2 * lane),
                0, 0);
#else
            ((float2*)dst)[lane] = ((const float2*)src)[lane];
#endif
        } else {
            ((float2*)dst)[lane] = make_float2(0.0f, 0.0f);
        }
    }
#if HAVE_ASYNC_LDS
#if __has_builtin(__builtin_amdgcn_s_wait_asynccnt)
    __builtin_amdgcn_s_wait_asynccnt(0);
#else
    asm volatile("s_wait_asynccnt 0x0" ::: "memory");
#endif
#endif
    __syncthreads();

    // ---- P1: per-(t,c) mean / unbiased var over k2 for X1 and X2 ----
    for (int q = tid; q < TT * C; q += 128) {
        const int t = q >> 6, c = q & 63;
        float s1 = 0.f, q1 = 0.f, s2 = 0.f, q2 = 0.f;
#pragma unroll
        for (int k = 0; k < K2; ++k) {
            float a = sX1[(t * K2 + k) * C + c]; s1 += a; q1 += a * a;
            float d = sX2[(t * K2 + k) * C + c]; s2 += d; q2 += d * d;
        }
        sM1[q] = s1 * (1.0f / 9.0f);
        sV1[q] = (q1 - s1 * s1 * (1.0f / 9.0f)) * (1.0f / 8.0f);
        sM2[q] = s2 * (1.0f / 9.0f);
        sV2[q] = (q2 - s2 * s2 * (1.0f / 9.0f)) * (1.0f / 8.0f);
    }
    __syncthreads();

    // ---- P2: 4x4 raw scores (mean & var attentions) ----
    if (tid < 16) {
        const int t = tid >> 2, t2 = tid & 3;
        float d3 = 0.f, d2 = 0.f;
        for (int c = 0; c < C; ++c) {
            d3 += sM1[t * C + c] * sM2[t2 * C + c];
            d2 += sV1[t * C + c] * sV2[t2 * C + c];
        }
        sS3[tid] = d3; sS2[tid] = d2;
    }
    __syncthreads();

    // ---- P3: softmax each 4-row, combine: w23 = softmax(s3) + softmax(s2) ----
    if (tid < TT) {
        float r3[4], r2[4], m3 = -1e30f, m2 = -1e30f;
#pragma unroll
        for (int j = 0; j < 4; ++j) {
            r3[j] = sS3[tid * 4 + j]; m3 = fmaxf(m3, r3[j]);
            r2[j] = sS2[tid * 4 + j]; m2 = fmaxf(m2, r2[j]);
        }
        float z3 = 0.f, z2 = 0.f;
#pragma unroll
        for (int j = 0; j < 4; ++j) {
            r3[j] = expf(r3[j] - m3); z3 += r3[j];
            r2[j] = expf(r2[j] - m2); z2 += r2[j];
        }
#pragma unroll
        for (int j = 0; j < 4; ++j)
            sW23[tid * 4 + j] = r3[j] / z3 + r2[j] / z2;
    }
    __syncthreads();

    // ---- P4: score1 raw = X1(36x64) @ X2^T via WMMA, waves 0..2 own M-tiles ----
    if (widx < 3) {
        const int r0 = widx * 16;
        v8f c0 = {}; v8f c1 = {}; v8f c2 = {};
#pragma unroll
        for (int k = 0; k < 16; ++k) {              // K = 64
            const int cb = 4 * k + 2 * half;
            v2f a  = *(const v2f*)(sX1 + (r0 + ln) * C + cb);
            v2f b0 = *(const v2f*)(sX2 + ( 0 + ln) * C + cb);  // B[kk][n]=X2[n][kk]
            v2f b1 = *(const v2f*)(sX2 + (16 + ln) * C + cb);
            v2f b2 = *(const v2f*)(sX2 + (32 + ln) * C + cb);
            c0 = __builtin_amdgcn_wmma_f32_16x16x4_f32(false, a, false, b0, (short)0, c0, false, false);
            c1 = __builtin_amdgcn_wmma_f32_16x16x4_f32(false, a, false, b1, (short)0, c1, false, false);
            c2 = __builtin_amdgcn_wmma_f32_16x16x4_f32(false, a, false, b2, (short)0, c2, false, false);
        }
#pragma unroll
        for (int r = 0; r < 8; ++r) {
            const int mrow = r0 + r + 8 * half;
            sS1[mrow * TKP +  0 + ln] = c0[r];
            sS1[mrow * TKP + 16 + ln] = c1[r];
            sS1[mrow * TKP + 32 + ln] = c2[r];
        }
    }
    __syncthreads();

    // ---- P5: softmax score1 rows over 36 cols (cols 36..47 stay exact 0) ----
    if (tid < TK) {
        float* row = sS1 + tid * TKP;
        float m = -1e30f;
        for (int j = 0; j < TK; ++j) m = fmaxf(m, row[j]);
        float z = 0.f;
        for (int j = 0; j < TK; ++j) { float e = expf(row[j] - m); row[j] = e; z += e; }
        const float inv = 1.0f / z;
        for (int j = 0; j < TK; ++j) row[j] *= inv;
    }
    __syncthreads();

    // ---- P6: x5 = score1(36x48pad) @ X3(48x64) via WMMA -> sX6 ----
    if (widx < 3) {
        const int r0 = widx * 16;
        v8f d0 = {}; v8f d1 = {}; v8f d2 = {}; v8f d3 = {};
#pragma unroll
        for (int k = 0; k < 12; ++k) {              // K = 48 (padded, zeros beyond 36)
            const int kb = 4 * k + 2 * half;
            v2f a = *(const v2f*)(sS1 + (r0 + ln) * TKP + kb);
            v2f b0 = { sX3[kb * C +  0 + ln], sX3[(kb + 1) * C +  0 + ln] };
            v2f b1 = { sX3[kb * C + 16 + ln], sX3[(kb + 1) * C + 16 + ln] };
            v2f b2 = { sX3[kb * C + 32 + ln], sX3[(kb + 1) * C + 32 + ln] };
            v2f b3 = { sX3[kb * C + 48 + ln], sX3[(kb + 1) * C + 48 + ln] };
            d0 = __builtin_amdgcn_wmma_f32_16x16x4_f32(false, a, false, b0, (short)0, d0, false, false);
            d1 = __builtin_amdgcn_wmma_f32_16x16x4_f32(false, a, false, b1, (short)0, d1, false, false);
            d2 = __builtin_amdgcn_wmma_f32_16x16x4_f32(false, a, false, b2, (short)0, d2, false, false);
            d3 = __builtin_amdgcn_wmma_f32_16x16x4_f32(false, a, false, b3, (short)0, d3, false, false);
        }
#pragma unroll
        for (int r = 0; r < 8; ++r) {
            const int mrow = r0 + r + 8 * half;
            if (mrow < TK) {
                sX6[mrow * C +  0 + ln] = d0[r];
                sX6[mrow * C + 16 + ln] = d1[r];
                sX6[mrow * C + 32 + ln] = d2[r];
                sX6[mrow * C + 48 + ln] = d3[r];
            }
        }
    }
    __syncthreads();

    // ---- P7: x6 = x5 + x4,  x4[t,k,c] = sum_t' w23[t][t'] * X3[t'*9+k][c] ----
    for (int e = tid; e < TK * C; e += 128) {
        const int row = e >> 6;
        const int c = e & 63;
        const int t = row / K2, k = row % K2;
        float v = sX6[e];
#pragma unroll
        for (int t2 = 0; t2 < TT; ++t2)
            v += sW23[t * 4 + t2] * sX3[(t2 * K2 + k) * C + c];
        sX6[e] = v;
    }
    __syncthreads();

    // ---- P8: dot with center (k2/2 == 4) ----
    if (tid < TK) {
        const int t = tid / K2;
        const float* xr = sX6 + tid * C;
        const float* cr = sX6 + (t * K2 + 4) * C;
        float d = 0.f;
        for (int c = 0; c < C; ++c) d += xr[c] * cr[c];
        sDot[tid] = d;
    }
    __syncthreads();

    // ---- P9: softmax over k2 ----
    if (tid < TT) {
        float m = -1e30f;
#pragma unroll
        for (int k = 0; k < K2; ++k) m = fmaxf(m, sDot[tid * K2 + k]);
        float e2[K2]; float z = 0.f;
#pragma unroll
        for (int k = 0; k < K2; ++k) { e2[k] = expf(sDot[tid * K2 + k] - m); z += e2[k]; }
        const float inv = 1.0f / z;
#pragma unroll
        for (int k = 0; k < K2; ++k) sSm4[tid * K2 + k] = e2[k] * inv;
    }
    __syncthreads();

    // ---- P10: out[n,t,c,h,w] = sum_k x6[t,k,c] * sm4[t,k] ----
    for (int e = tid; e < TT * C; e += 128) {
        const int t = e >> 6, c = e & 63;
        float o = 0.f;
#pragma unroll
        for (int k = 0; k < K2; ++k)
            o += sX6[(t * K2 + k) * C + c] * sSm4[t * K2 + k];
        out[((size_t)(nn * TT + t) * C + c) * HW + p] = o;
    }
}

extern "C" void kernel_launch(void* const* d_in, const int* in_sizes, int n_in,
                              void* d_out, int out_size, void* d_ws, size_t ws_size,
                              hipStream_t stream)
{
    const float* x  = (const float*)d_in[0];
    const float* w1 = (const float*)d_in[1];
    const float* w2 = (const float*)d_in[2];
    const float* w3 = (const float*)d_in[3];
    float* y   = (float*)d_ws;     // 3 * NT * HW * C floats = 37.7 MB scratch
    float* out = (float*)d_out;

    // 12288 waves / 8 per block = 1536 blocks
    stgn_conv_kernel<<<(NT * 4 * (HW / 16)) / 8, 256, 0, stream>>>(x, w1, w2, w3, y);
    // one pixel per block
    stgn_attn_kernel<<<NB * HW, 128, 0, stream>>>(y, out);
}